// GIN_Net_55061480734898
// MI455X (gfx1250) — compile-verified
//
#include <hip/hip_runtime.h>

#define NNODES 100000
#define NEDGES 1600000
#define HID 64
#define NTILES (NNODES / 16) /* 6250, exact */
#define BN_EPS 1e-5f

typedef __attribute__((ext_vector_type(2))) float v2f;
typedef __attribute__((ext_vector_type(8))) float v8f;

// ---------------------------------------------------------------- zero fill
__global__ void zero_kernel(float* __restrict__ p, int n) {
  int i = blockIdx.x * blockDim.x + threadIdx.x;
  if (i < n) p[i] = 0.0f;
}

// ------------------------------------------------- edge scatter (segment_sum)
// agg[dst] += x[src]; one thread per (edge, 4-float group); L2-resident table.
__global__ void scatter_kernel(const float* __restrict__ X,
                               const long long* __restrict__ ei,
                               float* __restrict__ AGG) {
  long long idx = (long long)blockIdx.x * blockDim.x + threadIdx.x;
  if (idx >= (long long)NEDGES * 16) return;
  int e = (int)(idx >> 4);
  int q = (int)(idx & 15);
  int src = (int)ei[e];
  int dst = (int)ei[NEDGES + e];
  float4 v = ((const float4*)(X + (size_t)src * HID))[q];
  float* d = AGG + (size_t)dst * HID + (size_t)q * 4;
  atomicAdd(d + 0, v.x);
  atomicAdd(d + 1, v.y);
  atomicAdd(d + 2, v.z);
  atomicAdd(d + 3, v.w);
}

// ------------------------------------------------------------ W -> LDS stage
// Repack W (64x64 row-major) into K-pair form so a B fragment
// {W[k][n], W[k+1][n]} is one contiguous float2 -> single ds_load_b64.
// Wl[p*64 + n] = {W[2p][n], W[2p+1][n]},  p in [0,32).
// Global side is fully coalesced (consecutive lanes -> consecutive n).
__device__ __forceinline__ void stage_W_lds(float2* __restrict__ Wl,
                                            const float* __restrict__ W) {
  for (int i = threadIdx.x; i < 32 * HID; i += blockDim.x) {
    int p = i >> 6;
    int n = i & 63;
    float2 t;
    t.x = W[(size_t)(2 * p) * HID + n];
    t.y = W[(size_t)(2 * p + 1) * HID + n];
    Wl[i] = t;
  }
}

// -------------------------------------------------------------- WMMA helper
// 4 N-tiles of a 16-wide output strip; A frag shared, B frags from LDS.
// Layouts per CDNA5 ISA 7.12.2: A 16x4 f32 -> lane m=l&15, K pair 2*(l>>4);
// B 4x16 mirrored; C/D: M = vgpr + 8*(l>>4), N = l&15.
__device__ __forceinline__ void wmma_accum4(v8f c[4], v2f a,
                                            const float2* __restrict__ Wl,
                                            int cidx, int n) {
  const float2* row = Wl + (size_t)(cidx >> 1) * HID;  // cidx always even
  float2 t;
  v2f b;
  t = row[n];      b.x = t.x; b.y = t.y;
  c[0] = __builtin_amdgcn_wmma_f32_16x16x4_f32(false, a, false, b, (short)0, c[0], false, false);
  t = row[n + 16]; b.x = t.x; b.y = t.y;
  c[1] = __builtin_amdgcn_wmma_f32_16x16x4_f32(false, a, false, b, (short)0, c[1], false, false);
  t = row[n + 32]; b.x = t.x; b.y = t.y;
  c[2] = __builtin_amdgcn_wmma_f32_16x16x4_f32(false, a, false, b, (short)0, c[2], false, false);
  t = row[n + 48]; b.x = t.x; b.y = t.y;
  c[3] = __builtin_amdgcn_wmma_f32_16x16x4_f32(false, a, false, b, (short)0, c[3], false, false);
}

// ------------------------------------------ hpre = (x + agg) @ W1 + b1
__global__ void gemm_xplusagg_kernel(const float* __restrict__ X,
                                     const float* __restrict__ AGG,
                                     const float* __restrict__ W,
                                     const float* __restrict__ bias,
                                     float* __restrict__ OUT) {
  __shared__ float2 Wl[32 * HID];  // 16 KB
  stage_W_lds(Wl, W);
  __syncthreads();

  int lane = threadIdx.x & 31;
  int tile = blockIdx.x * (blockDim.x >> 5) + (threadIdx.x >> 5);
  if (tile < NTILES) {  // wave-uniform; EXEC stays all-1 for WMMA
    int m = lane & 15;
    int kh = lane >> 4;
    const float* xr = X   + (size_t)(tile * 16 + m) * HID;
    const float* gr = AGG + (size_t)(tile * 16 + m) * HID;
    v8f c[4] = {};
#pragma unroll
    for (int kk = 0; kk < 16; ++kk) {
      int cidx = 4 * kk + 2 * kh;
      v2f a;
      a.x = xr[cidx] + gr[cidx];
      a.y = xr[cidx + 1] + gr[cidx + 1];
      wmma_accum4(c, a, Wl, cidx, m);
    }
#pragma unroll
    for (int nt = 0; nt < 4; ++nt) {
      int col = nt * 16 + m;
      float bc = bias[col];
#pragma unroll
      for (int v = 0; v < 8; ++v) {
        int row = tile * 16 + v + 8 * kh;
        OUT[(size_t)row * HID + col] = c[nt][v] + bc;
      }
    }
  }
}

// ------------------------------------- column sum / sumsq for BatchNorm stats
__global__ void bn_stats_kernel(const float* __restrict__ H,
                                float* __restrict__ stats) {
  __shared__ float ssum[256];
  __shared__ float ssq[256];
  int tid = threadIdx.x;
  int col = tid & 63;
  float s = 0.0f, q = 0.0f;
  for (int r = blockIdx.x * 4 + (tid >> 6); r < NNODES; r += gridDim.x * 4) {
    float v = H[(size_t)r * HID + col];
    s += v;
    q += v * v;
  }
  ssum[tid] = s;
  ssq[tid] = q;
  __syncthreads();
  if (tid < 64) {
    s = ssum[tid] + ssum[tid + 64] + ssum[tid + 128] + ssum[tid + 192];
    q = ssq[tid] + ssq[tid + 64] + ssq[tid + 128] + ssq[tid + 192];
    atomicAdd(&stats[col], s);
    atomicAdd(&stats[64 + col], q);
  }
}

// -------------------- fold gamma/beta/mean/var into per-column scale & shift
__global__ void bn_final_kernel(float* __restrict__ stats,
                                const float* __restrict__ gamma,
                                const float* __restrict__ beta) {
  int c = threadIdx.x;
  if (c < 64) {
    float inv_n = 1.0f / (float)NNODES;
    float mean = stats[c] * inv_n;
    float var = stats[64 + c] * inv_n - mean * mean;
    float sc = gamma[c] * rsqrtf(var + BN_EPS);
    stats[128 + c] = sc;                  // scale
    stats[192 + c] = beta[c] - mean * sc; // shift
  }
}

// --------------- h_out = relu( relu(scale*hpre + shift) @ W2 + b2 )  (fused)
__global__ void gemm_bn_kernel(const float* __restrict__ H,
                               const float* __restrict__ stats,
                               const float* __restrict__ W,
                               const float* __restrict__ bias,
                               float* __restrict__ OUT) {
  __shared__ float2 Wl[32 * HID];  // 16 KB
  stage_W_lds(Wl, W);
  __syncthreads();

  int lane = threadIdx.x & 31;
  int tile = blockIdx.x * (blockDim.x >> 5) + (threadIdx.x >> 5);
  if (tile < NTILES) {
    int m = lane & 15;
    int kh = lane >> 4;
    const float* hr = H + (size_t)(tile * 16 + m) * HID;
    const float* scale = stats + 128;
    const float* shift = stats + 192;
    v8f c[4] = {};
#pragma unroll
    for (int kk = 0; kk < 16; ++kk) {
      int cidx = 4 * kk + 2 * kh;
      v2f a;
      a.x = fmaxf(scale[cidx] * hr[cidx] + shift[cidx], 0.0f);
      a.y = fmaxf(scale[cidx + 1] * hr[cidx + 1] + shift[cidx + 1], 0.0f);
      wmma_accum4(c, a, Wl, cidx, m);
    }
#pragma unroll
    for (int nt = 0; nt < 4; ++nt) {
      int col = nt * 16 + m;
      float bc = bias[col];
#pragma unroll
      for (int v = 0; v < 8; ++v) {
        int row = tile * 16 + v + 8 * kh;
        OUT[(size_t)row * HID + col] = fmaxf(c[nt][v] + bc, 0.0f);  // outer relu
      }
    }
  }
}

// ---------------- out = [x | h1 | h2] @ Wjk + bjk   (three 64x64 blocks)
__global__ void gemm_jk_kernel(const float* __restrict__ X,
                               const float* __restrict__ H1,
                               const float* __restrict__ H2,
                               const float* __restrict__ Wjk,
                               const float* __restrict__ bias,
                               float* __restrict__ OUT) {
  __shared__ float2 Wl[3 * 32 * HID];  // 48 KB (three 64x64 blocks of Wjk)
  stage_W_lds(Wl,               Wjk);
  stage_W_lds(Wl + 32 * HID,    Wjk + (size_t)HID * HID);
  stage_W_lds(Wl + 2 * 32 * HID, Wjk + (size_t)2 * HID * HID);
  __syncthreads();

  int lane = threadIdx.x & 31;
  int tile = blockIdx.x * (blockDim.x >> 5) + (threadIdx.x >> 5);
  if (tile < NTILES) {
    int m = lane & 15;
    int kh = lane >> 4;
    const float* srcs[3] = {X, H1, H2};
    v8f c[4] = {};
#pragma unroll
    for (int s = 0; s < 3; ++s) {
      const float* ar = srcs[s] + (size_t)(tile * 16 + m) * HID;
      const float2* Wb = Wl + (size_t)s * 32 * HID;
#pragma unroll
      for (int kk = 0; kk < 16; ++kk) {
        int cidx = 4 * kk + 2 * kh;
        v2f a;
        a.x = ar[cidx];
        a.y = ar[cidx + 1];
        wmma_accum4(c, a, Wb, cidx, m);
      }
    }
#pragma unroll
    for (int nt = 0; nt < 4; ++nt) {
      int col = nt * 16 + m;
      float bc = bias[col];
#pragma unroll
      for (int v = 0; v < 8; ++v) {
        int row = tile * 16 + v + 8 * kh;
        OUT[(size_t)row * HID + col] = c[nt][v] + bc;
      }
    }
  }
}

// ---------------------------------------------------------------- launcher
extern "C" void kernel_launch(void* const* d_in, const int* in_sizes, int n_in,
                              void* d_out, int out_size, void* d_ws, size_t ws_size,
                              hipStream_t stream) {
  (void)in_sizes; (void)n_in; (void)out_size; (void)ws_size;

  const float* x = (const float*)d_in[0];
  const long long* ei = (const long long*)d_in[1];  // int64 edge_index
  // d_in[2] edge_attr: unused by GINConv
  const float* W1[2] = {(const float*)d_in[3], (const float*)d_in[9]};
  const float* b1[2] = {(const float*)d_in[4], (const float*)d_in[10]};
  const float* gm[2] = {(const float*)d_in[5], (const float*)d_in[11]};
  const float* bt[2] = {(const float*)d_in[6], (const float*)d_in[12]};
  const float* W2[2] = {(const float*)d_in[7], (const float*)d_in[13]};
  const float* b2[2] = {(const float*)d_in[8], (const float*)d_in[14]};
  const float* Wjk = (const float*)d_in[15];
  const float* bjk = (const float*)d_in[16];

  float* ws = (float*)d_ws;
  const size_t NF = (size_t)NNODES * HID;
  float* agg   = ws;
  float* hpre  = ws + NF;
  float* h1    = ws + 2 * NF;
  float* h2    = ws + 3 * NF;
  float* stats = ws + 4 * NF;  // 256 floats: sum|sumsq|scale|shift

  const int zgrid = (int)((NF + 255) / 256);
  const int sgrid = (int)(((long long)NEDGES * 16 + 255) / 256);
  const int ggrid = (NTILES + 7) / 8;  // 8 waves per 256-thread block

  const float* lin = x;
  float* lout[2] = {h1, h2};
  for (int L = 0; L < 2; ++L) {
    zero_kernel<<<zgrid, 256, 0, stream>>>(agg, (int)NF);
    zero_kernel<<<1, 256, 0, stream>>>(stats, 256);
    scatter_kernel<<<sgrid, 256, 0, stream>>>(lin, ei, agg);
    gemm_xplusagg_kernel<<<ggrid, 256, 0, stream>>>(lin, agg, W1[L], b1[L], hpre);
    bn_stats_kernel<<<256, 256, 0, stream>>>(hpre, stats);
    bn_final_kernel<<<1, 64, 0, stream>>>(stats, gm[L], bt[L]);
    gemm_bn_kernel<<<ggrid, 256, 0, stream>>>(hpre, stats, W2[L], b2[L], lout[L]);
    lin = lout[L];
  }
  gemm_jk_kernel<<<ggrid, 256, 0, stream>>>(x, h1, h2, Wjk, bjk, (float*)d_out);
}